// HHNet_No_Gating_62397284876291
// MI455X (gfx1250) — compile-verified
//
#include <hip/hip_runtime.h>
#include <stdint.h>

typedef __attribute__((ext_vector_type(2))) float v2f;
typedef __attribute__((ext_vector_type(8))) float v8f;

#define B_SZ 100
#define T_SZ 2000
#define F_SZ 784
#define C1   100
#define C2   10
#define LDS_STRIDE 788   // 788 mod 64 = 20 -> conflict-free b64 LDS reads across 16 rows

#if defined(__has_builtin)
#if __has_builtin(__builtin_amdgcn_global_load_async_to_lds_b128)
#define HAVE_ASYNC_LDS 1
#endif
#endif

#if HAVE_ASYNC_LDS
typedef int v4i __attribute__((vector_size(4 * sizeof(int))));
typedef __attribute__((address_space(1))) v4i ga_v4i;   // global
typedef __attribute__((address_space(3))) v4i ls_v4i;   // LDS
#endif

// ---------- W1 repack: W1 (100x784) -> k-pair-major w1p -------------------
// w1p[(k/2)*200 + 2*c + (k&1)] = W1[c][k]  => B operand is a contiguous b64.
__global__ __launch_bounds__(256) void repack_w1(const float* __restrict__ w1,
                                                 float* __restrict__ w1p) {
    int i = blockIdx.x * 256 + threadIdx.x;
    if (i < C1 * F_SZ) {
        int c = i / F_SZ;
        int k = i - c * F_SZ;
        w1p[(k >> 1) * (2 * C1) + (c << 1) + (k & 1)] = w1[i];
    }
}

// ---------- z1 = batch @ W1^T via V_WMMA_F32_16X16X4_F32 ------------------
// grid.x = 200000/16 = 12500 M-tiles; 7 waves per block, one 16-col N-tile each.
__global__ __launch_bounds__(224) void gemm_z1(const float* __restrict__ batch,
                                               const float* __restrict__ w1p,
                                               float* __restrict__ z1) {
    __shared__ float As[16 * LDS_STRIDE];
    const int m0  = blockIdx.x * 16;
    const int tid = threadIdx.x;

    // Stage the 16x784 A panel into LDS: 3136 16-byte chunks / 224 threads = 14 each.
#if HAVE_ASYNC_LDS
    for (int idx = tid; idx < 16 * (F_SZ / 4); idx += 224) {
        int row = idx / (F_SZ / 4);
        int c4  = idx - row * (F_SZ / 4);
        const float* g = batch + (size_t)(m0 + row) * F_SZ + c4 * 4;
        float*       l = &As[row * LDS_STRIDE + c4 * 4];
        __builtin_amdgcn_global_load_async_to_lds_b128(
            (ga_v4i*)(uintptr_t)g, (ls_v4i*)(uintptr_t)l, 0, 0);
    }
    asm volatile("s_wait_asynccnt 0x0" ::: "memory");   // barrier does not cover ASYNCcnt
#else
    for (int idx = tid; idx < 16 * (F_SZ / 4); idx += 224) {
        int row = idx / (F_SZ / 4);
        int c4  = idx - row * (F_SZ / 4);
        const float4 v = ((const float4*)(batch + (size_t)(m0 + row) * F_SZ))[c4];
        *((float4*)&As[row * LDS_STRIDE + c4 * 4]) = v;
    }
#endif
    __syncthreads();

    const int wave = tid >> 5;          // N-tile id 0..6
    const int lane = tid & 31;
    const int nn   = lane & 15;
    const int hh   = lane >> 4;
    const int col  = wave * 16 + nn;
    const int colc = (col < C1) ? col : (C1 - 1);   // clamp loads for tail tile

    v8f acc = {};
    const float* arow = &As[(lane & 15) * LDS_STRIDE + 2 * hh];

    for (int kk = 0; kk < F_SZ; kk += 4) {
        // A 16x4 fp32 layout: lane holds {A[m][kk+2h], A[m][kk+2h+1]}
        v2f a = *(const v2f*)(arow + kk);
        // B 4x16 fp32 layout: lane holds {W1[c][kk+2h], W1[c][kk+2h+1]} = one b64
        v2f b = *(const v2f*)(w1p + ((kk + 2 * hh) >> 1) * (2 * C1) + 2 * colc);
        acc = __builtin_amdgcn_wmma_f32_16x16x4_f32(false, a, false, b,
                                                    (short)0, acc, false, false);
    }

    if (col < C1) {
        #pragma unroll
        for (int v = 0; v < 8; ++v) {
            int mrow = m0 + v + 8 * hh;          // C/D layout: VGPR v -> row v+8h
            z1[(size_t)mrow * C1 + col] = acc[v];
        }
    }
}

// ---------------- HH dynamics (fused two-layer scan) ----------------
__device__ __forceinline__ float sigmoidf(float x) {
    return __fdividef(1.f, 1.f + __expf(-x));
}
__device__ __forceinline__ float gate_update(float a, float b, float g) {
    float s = 0.005f * (a + b);                      // 0.5*DT*(a+b)
    return __fdividef(a * 0.01f + (1.f - s) * g, s + 1.f);
}
__device__ __forceinline__ void hh_step(float zk, float& V, float& m, float& n, float& h) {
    float pow1 = 40.f * m * m * m * h;               // gna*m^3*h
    float n2   = n * n;
    float pow2 = 35.f * n2 * n2;                     // gk*n^4
    float Gs = 0.005f * (pow1 + pow2 + 0.3f);
    float E  = pow1 * 55.f + pow2 * (-77.f) + 0.3f * (-65.f);
    float Vn = __fdividef(V * (1.f - Gs) + 0.01f * (E + 1.5f + zk), 1.f + Gs);
    bool e25  = (Vn == 25.f);
    bool em35 = (Vn == -35.f);
    float V25 = e25  ? 0.f : Vn;
    float V35 = em35 ? 0.f : Vn;
    float aN = __fdividef( 0.02f  * (V25 - 25.f), 1.f - __expf((25.f - V25) * (1.f / 9.f)));
    float bN = __fdividef(-0.002f * (V25 - 25.f), 1.f - __expf((V25 - 25.f) * (1.f / 9.f)));
    float aM = __fdividef( 0.182f * (V35 + 35.f), 1.f - __expf((-V35 - 35.f) * (1.f / 9.f)));
    float bM = __fdividef(-0.124f * (V35 + 35.f), 1.f - __expf((V35 + 35.f) * (1.f / 9.f)));
    float aH = 0.25f * __expf((-Vn - 90.f) * (1.f / 12.f));
    float bH = 0.25f * __expf((Vn + 34.f) * (1.f / 12.f));
    if (e25)  { aN = 0.18f;  bN = 0.018f; }
    if (em35) { aM = 1.638f; bM = 1.116f; }
    m = gate_update(aM, bM, m);
    n = gate_update(aN, bN, n);
    h = gate_update(aH, bH, h);
    V = Vn;
}

// One block per batch element b. 160 threads = 5 waves:
//   tids   0..99  : layer-1 neurons (waves 0..3) + z2 partial dots
//   tids 128..137 : layer-2 neurons (wave 4) -- runs concurrently with the
//                   layer-1 HH step after the second barrier.
__global__ __launch_bounds__(160) void hh_fused(const float* __restrict__ z1,
                                                const float* __restrict__ W2,
                                                float* __restrict__ out) {
    const int b   = blockIdx.x;
    const int tid = threadIdx.x;
    const int l2  = tid - 128;            // layer-2 neuron id for wave 4
    __shared__ float T1s[C1];
    __shared__ float zp[C2][11];          // padded to dodge bank conflicts

    const int o  = tid / 10;              // reduction role for tids 0..99
    const int ch = tid - o * 10;
    float w2c[10];
    if (tid < C1) {
        #pragma unroll
        for (int j = 0; j < 10; ++j) w2c[j] = W2[o * C1 + ch * 10 + j];
    }

    float V1 = -70.f, m1 = 0.f, n1 = 0.f, h1 = 1.f;
    float V2 = -70.f, m2 = 0.f, n2 = 0.f, h2 = 1.f;

    float* outb = out + (size_t)b * T_SZ * C2;
    if (l2 >= 0 && l2 < C2)
        outb[l2] = sigmoidf((-70.f - 20.f) * (1.f / 3.f));   // t = 0

    const float* z1b = z1 + (size_t)b * T_SZ * C1;

    for (int t = 1; t < T_SZ; ++t) {
        float zk1 = 0.f;
        if (tid < C1) {
            zk1 = z1b[(t - 1) * C1 + tid];                  // issue load early
            T1s[tid] = sigmoidf((V1 - 20.f) * (1.f / 3.f)); // T1[b, t-1, tid]
        }
        __syncthreads();
        if (tid < C1) {                                     // partial dot: 10 MACs/lane
            float s = 0.f;
            #pragma unroll
            for (int j = 0; j < 10; ++j) s += T1s[ch * 10 + j] * w2c[j];
            zp[o][ch] = s;
        }
        __syncthreads();
        // Wave 4: finish z2 reduction, integrate layer 2, store output.
        if (l2 >= 0 && l2 < C2) {
            float z2 = 0.f;
            #pragma unroll
            for (int c = 0; c < 10; ++c) z2 += zp[l2][c];
            hh_step(z2, V2, m2, n2, h2);
            outb[t * C2 + l2] = sigmoidf((V2 - 20.f) * (1.f / 3.f));
        }
        // Waves 0..3: layer-1 HH step, concurrent with wave 4.
        if (tid < C1) hh_step(zk1, V1, m1, n1, h1);
        // next iteration's barrier pair protects T1s / zp reuse
    }
}

extern "C" void kernel_launch(void* const* d_in, const int* in_sizes, int n_in,
                              void* d_out, int out_size, void* d_ws, size_t ws_size,
                              hipStream_t stream) {
    const float* batch = (const float*)d_in[0];   // (100, 2000, 784) fp32
    const float* W1    = (const float*)d_in[1];   // (100, 784)
    const float* W2    = (const float*)d_in[2];   // (10, 100)
    float* out = (float*)d_out;                   // (100, 2000, 10)

    float* w1p = (float*)d_ws;                                  // 784*100 floats
    float* z1  = (float*)((char*)d_ws + (size_t)F_SZ * C1 * 4); // 200000*100 floats (80 MB)

    repack_w1<<<(C1 * F_SZ + 255) / 256, 256, 0, stream>>>(W1, w1p);
    gemm_z1<<<(B_SZ * T_SZ) / 16, 224, 0, stream>>>(batch, w1p, z1);
    hh_fused<<<B_SZ, 160, 0, stream>>>(z1, W2, out);
}